// GINBaseline_31739808318046
// MI455X (gfx1250) — compile-verified
//
#include <hip/hip_runtime.h>

#define NNODES   100000
#define NEDGES   3200000
#define DIM      128
#define NGRAPHS  512
#define NCLASSES 10
#define NLAYERS  3

typedef float v2f __attribute__((ext_vector_type(2)));
typedef float v8f __attribute__((ext_vector_type(8)));

// LDS weight layout: paired along K so a B fragment {W[kk][n], W[kk+1][n]}
// is one aligned 8-byte word -> single ds_load_b64 straight into an even
// VGPR pair (no repacking movs before v_wmma).
//   Wl[(k>>1)*(2*DIM) + n*2 + (k&1)] = W[k][n]
__device__ __forceinline__ int wswz(int k, int n) {
    return (k >> 1) * (2 * DIM) + n * 2 + (k & 1);
}

// ---------------------------------------------------------------------------
// Encoder GEMM: out[M x 128] = A[M x 128] @ W[128 x 128] + bias
// One wave = one 16-row tile, full 128 output columns (8 f32 accumulators).
// ---------------------------------------------------------------------------
__global__ __launch_bounds__(256) void wmma_encoder_kernel(
    const float* __restrict__ A, const float* __restrict__ W,
    const float* __restrict__ bias, float* __restrict__ out, int M)
{
    extern __shared__ float lds[];
    float* Wl = lds;             // 128*128 floats, paired-K layout
    float* bl = lds + DIM * DIM; // 128 floats
    const int tid = threadIdx.x;

    for (int i = tid; i < DIM * DIM; i += 256) {
        const int k = i >> 7, n = i & (DIM - 1);
        Wl[wswz(k, n)] = W[i];
    }
    if (tid < DIM) bl[tid] = bias[tid];
    __syncthreads();

    const int wave  = tid >> 5;
    const int lane  = tid & 31;
    const int lrow  = lane & 15;   // row (A) / col (B) within tile
    const int lhalf = lane >> 4;   // K-half selector
    const int rowTile = blockIdx.x * 128 + wave * 16;
    const int arow = rowTile + lrow;
    // Clamp instead of predicating: padding A-rows only affect padding C-rows,
    // which are never stored. Keeps EXEC full / loop branch-free.
    const int arowc = arow < M ? arow : (M - 1);
    const float* Ap = A + (size_t)arowc * DIM;

    v8f acc[8] = {};

    for (int k = 0; k < DIM; k += 4) {
        const int kk = k + 2 * lhalf;
        const v2f a = *(const v2f*)(Ap + kk);          // A: M=lane%16, K=kk..kk+1
        const float* Bp = Wl + (kk >> 1) * (2 * DIM) + lrow * 2;
        #pragma unroll
        for (int t = 0; t < 8; ++t) {
            const v2f b = *(const v2f*)(Bp + t * 32);  // ds_load_b64, pre-paired
            acc[t] = __builtin_amdgcn_wmma_f32_16x16x4_f32(
                false, a, false, b, (short)0, acc[t], false, false);
        }
    }

    #pragma unroll
    for (int t = 0; t < 8; ++t) {
        const int col = t * 16 + lrow;
        const float bv = bl[col];
        #pragma unroll
        for (int r = 0; r < 8; ++r) {
            const int orow = rowTile + r + 8 * lhalf;  // C: M = r + 8*half
            if (orow < M) out[(size_t)orow * DIM + col] = acc[t][r] + bv;
        }
    }
}

// ---------------------------------------------------------------------------
// Edge scatter: agg[dst[e]] += h[src[e]]  (wave per edge, float4 per lane,
// L2-resident atomics)
// ---------------------------------------------------------------------------
__global__ __launch_bounds__(256) void scatter_add_kernel(
    const float* __restrict__ h, const int* __restrict__ src,
    const int* __restrict__ dst, float* __restrict__ agg, int E)
{
    const int e = blockIdx.x * 8 + (threadIdx.x >> 5);
    if (e >= E) return;
    const int lane = threadIdx.x & 31;
    const int s = src[e];
    const int d = dst[e];
    const float4 v = *(const float4*)(h + (size_t)s * DIM + lane * 4);
    float* ap = agg + (size_t)d * DIM + lane * 4;
    atomicAdd(ap + 0, v.x);
    atomicAdd(ap + 1, v.y);
    atomicAdd(ap + 2, v.z);
    atomicAdd(ap + 3, v.w);
}

// ---------------------------------------------------------------------------
// Fused GIN MLP (per layer), in place on h:
//   z = h + agg ; t = relu(z @ W1 + b1) ; h = relu(t @ W2 + b2)
// LDS: weights 64KB (W1 reloaded with W2) + per-wave 16x128 z/t tile (64KB).
// ---------------------------------------------------------------------------
__global__ __launch_bounds__(256) void gin_mlp_kernel(
    float* __restrict__ h, const float* __restrict__ agg,
    const float* __restrict__ W1, const float* __restrict__ b1,
    const float* __restrict__ W2, const float* __restrict__ b2, int M)
{
    extern __shared__ float lds[];
    float* Wl = lds;                   // 16384 floats, paired-K layout
    float* bl = lds + DIM * DIM;       // 128 floats
    float* tl = bl + DIM;              // 8 waves * 16 * 128 floats

    const int tid  = threadIdx.x;
    const int wave = tid >> 5;
    const int lane = tid & 31;
    const int lrow  = lane & 15;
    const int lhalf = lane >> 4;
    const int rowTile = blockIdx.x * 128 + wave * 16;
    float* zt = tl + wave * (16 * DIM);   // this wave's 16x128 tile

    // Stage W1/b1 + coalesced z = h + agg tile.
    for (int i = tid; i < DIM * DIM; i += 256) {
        const int k = i >> 7, n = i & (DIM - 1);
        Wl[wswz(k, n)] = W1[i];
    }
    if (tid < DIM) bl[tid] = b1[tid];
    for (int r = 0; r < 16; ++r) {
        const int grow = rowTile + r;
        const int growc = grow < M ? grow : (M - 1);   // clamp, no divergence
        const float4 hv = *(const float4*)(h   + (size_t)growc * DIM + lane * 4);
        const float4 av = *(const float4*)(agg + (size_t)growc * DIM + lane * 4);
        float4 z;
        z.x = hv.x + av.x; z.y = hv.y + av.y;
        z.z = hv.z + av.z; z.w = hv.w + av.w;
        *(float4*)(zt + r * DIM + lane * 4) = z;
    }
    __syncthreads();

    // GEMM 1: t = relu(z @ W1 + b1)
    v8f acc[8] = {};
    for (int k = 0; k < DIM; k += 4) {
        const int kk = k + 2 * lhalf;
        const v2f a = *(const v2f*)(zt + lrow * DIM + kk);   // ds_load_b64
        const float* Bp = Wl + (kk >> 1) * (2 * DIM) + lrow * 2;
        #pragma unroll
        for (int t = 0; t < 8; ++t) {
            const v2f b = *(const v2f*)(Bp + t * 32);
            acc[t] = __builtin_amdgcn_wmma_f32_16x16x4_f32(
                false, a, false, b, (short)0, acc[t], false, false);
        }
    }
    // bias + relu, overwrite this wave's tile with t (wave-local; the
    // compiler's dscnt waits cover the RAW on zt within the wave)
    #pragma unroll
    for (int t = 0; t < 8; ++t) {
        const int col = t * 16 + lrow;
        const float bv = bl[col];
        #pragma unroll
        for (int r = 0; r < 8; ++r)
            zt[(r + 8 * lhalf) * DIM + col] = fmaxf(acc[t][r] + bv, 0.f);
    }

    // Swap weights to W2/b2 under full barriers.
    __syncthreads();
    for (int i = tid; i < DIM * DIM; i += 256) {
        const int k = i >> 7, n = i & (DIM - 1);
        Wl[wswz(k, n)] = W2[i];
    }
    if (tid < DIM) bl[tid] = b2[tid];
    __syncthreads();

    // GEMM 2: h = relu(t @ W2 + b2), in place (block only touches own rows).
    v8f acc2[8] = {};
    for (int k = 0; k < DIM; k += 4) {
        const int kk = k + 2 * lhalf;
        const v2f a = *(const v2f*)(zt + lrow * DIM + kk);
        const float* Bp = Wl + (kk >> 1) * (2 * DIM) + lrow * 2;
        #pragma unroll
        for (int t = 0; t < 8; ++t) {
            const v2f b = *(const v2f*)(Bp + t * 32);
            acc2[t] = __builtin_amdgcn_wmma_f32_16x16x4_f32(
                false, a, false, b, (short)0, acc2[t], false, false);
        }
    }
    #pragma unroll
    for (int t = 0; t < 8; ++t) {
        const int col = t * 16 + lrow;
        const float bv = bl[col];
        #pragma unroll
        for (int r = 0; r < 8; ++r) {
            const int orow = rowTile + r + 8 * lhalf;
            if (orow < M)
                h[(size_t)orow * DIM + col] = fmaxf(acc2[t][r] + bv, 0.f);
        }
    }
}

// ---------------------------------------------------------------------------
// Global add pool: g[batch[n]] += h[n]   (wave per node)
// ---------------------------------------------------------------------------
__global__ __launch_bounds__(256) void pool_kernel(
    const float* __restrict__ h, const int* __restrict__ batch,
    float* __restrict__ g, int N)
{
    const int n = blockIdx.x * 8 + (threadIdx.x >> 5);
    if (n >= N) return;
    const int lane = threadIdx.x & 31;
    const int b = batch[n];
    const float4 v = *(const float4*)(h + (size_t)n * DIM + lane * 4);
    float* gp = g + (size_t)b * DIM + lane * 4;
    atomicAdd(gp + 0, v.x);
    atomicAdd(gp + 1, v.y);
    atomicAdd(gp + 2, v.z);
    atomicAdd(gp + 3, v.w);
}

// ---------------------------------------------------------------------------
// Readout: out = relu(g @ Wr1 + br1) @ Wr2 + br2   (one block per graph)
// ---------------------------------------------------------------------------
__global__ __launch_bounds__(128) void readout_kernel(
    const float* __restrict__ g, const float* __restrict__ Wr1,
    const float* __restrict__ br1, const float* __restrict__ Wr2,
    const float* __restrict__ br2, float* __restrict__ out)
{
    __shared__ float grow[DIM];
    __shared__ float trow[DIM];
    const int row = blockIdx.x;
    const int c = threadIdx.x;
    grow[c] = g[(size_t)row * DIM + c];
    __syncthreads();
    float s = br1[c];
    for (int k = 0; k < DIM; ++k) s += grow[k] * Wr1[k * DIM + c];
    trow[c] = fmaxf(s, 0.f);
    __syncthreads();
    if (c < NCLASSES) {
        float o = br2[c];
        for (int k = 0; k < DIM; ++k) o += trow[k] * Wr2[k * NCLASSES + c];
        out[(size_t)row * NCLASSES + c] = o;
    }
}

// ---------------------------------------------------------------------------
extern "C" void kernel_launch(void* const* d_in, const int* in_sizes, int n_in,
                              void* d_out, int out_size, void* d_ws, size_t ws_size,
                              hipStream_t stream)
{
    const float* x     = (const float*)d_in[0];
    const int*   src   = (const int*)  d_in[1];
    const int*   dst   = (const int*)  d_in[2];
    const int*   batch = (const int*)  d_in[3];
    const float* W_enc = (const float*)d_in[4];
    const float* b_enc = (const float*)d_in[5];
    const float* W1    = (const float*)d_in[6];   // [3,128,128]
    const float* b1    = (const float*)d_in[7];   // [3,128]
    const float* W2    = (const float*)d_in[8];
    const float* b2    = (const float*)d_in[9];
    const float* Wr1   = (const float*)d_in[10];
    const float* br1   = (const float*)d_in[11];
    const float* Wr2   = (const float*)d_in[12];
    const float* br2   = (const float*)d_in[13];
    float* out = (float*)d_out;

    // Workspace: h [N,D] | agg [N,D] | g [G,D]
    float* h   = (float*)d_ws;
    float* agg = h + (size_t)NNODES * DIM;
    float* g   = agg + (size_t)NNODES * DIM;

    const int gemmBlocks = (NNODES + 127) / 128;
    const size_t smemEnc = (size_t)(DIM * DIM + DIM) * sizeof(float);                // ~64.5 KB
    const size_t smemMlp = (size_t)(DIM * DIM + DIM + 8 * 16 * DIM) * sizeof(float); // ~128.5 KB

    // Encoder: h = x @ W_enc + b_enc
    wmma_encoder_kernel<<<gemmBlocks, 256, smemEnc, stream>>>(x, W_enc, b_enc, h, NNODES);

    for (int l = 0; l < NLAYERS; ++l) {
        hipMemsetAsync(agg, 0, (size_t)NNODES * DIM * sizeof(float), stream);
        scatter_add_kernel<<<(NEDGES + 7) / 8, 256, 0, stream>>>(h, src, dst, agg, NEDGES);
        gin_mlp_kernel<<<gemmBlocks, 256, smemMlp, stream>>>(
            h, agg, W1 + (size_t)l * DIM * DIM, b1 + (size_t)l * DIM,
            W2 + (size_t)l * DIM * DIM, b2 + (size_t)l * DIM, NNODES);
    }

    hipMemsetAsync(g, 0, (size_t)NGRAPHS * DIM * sizeof(float), stream);
    pool_kernel<<<(NNODES + 7) / 8, 256, 0, stream>>>(h, batch, g, NNODES);
    readout_kernel<<<NGRAPHS, 128, 0, stream>>>(g, Wr1, br1, Wr2, br2, out);
}